// CapsuleLayer_86260123173141
// MI455X (gfx1250) — compile-verified
//
#include <hip/hip_runtime.h>
#include <math.h>

typedef __attribute__((ext_vector_type(2))) float v2f;
typedef __attribute__((ext_vector_type(8))) float v8f;

#define B_DIM   64
#define N_CAPS  1152
#define IN_DIM  8
#define OUT_DIM 16
#define K_TOT   (N_CAPS * IN_DIM)          /* 9216 */
#define WAVES   32                         /* 1024 threads: 8 waves / SIMD32 */
#define K_PER_WAVE (K_TOT / WAVES)         /* 288 */
#define K_STEPS    (K_PER_WAVE / 4)        /* 72  */

// Kernel A: S[b,d] = sum_{n,k} x[b,n,k] * W[n,k,d] via V_WMMA_F32_16X16X4_F32
// (K flattened over (n,k)), then v[b,d] = squash(S[b,:]/N)[d].
// One block per 16-row tile of b; 32 waves split K; register-double-buffered
// fragment loads so WMMA issue is not serialized behind memory latency.
__global__ __launch_bounds__(1024)
void caps_reduce_kernel(const float* __restrict__ x,
                        const float* __restrict__ Wf,
                        float* __restrict__ v_out) {
    __shared__ float partials[WAVES][32][8];   // 32 KB: per-wave C fragments
    __shared__ float smat[16][17];             // S tile, padded

    const int tid    = threadIdx.x;
    const int lane   = tid & 31;
    const int wave   = tid >> 5;
    const int mtile  = blockIdx.x;             // 0..3 -> rows b = 16*mtile ..
    const int laneLo = lane & 15;
    const int hiSel  = lane >> 4;              // 0: K pair {0,1}, 1: K pair {2,3}

    // A-matrix: lane holds A[M = brow][K = ksel, ksel+1] per ISA 16x4 f32 layout.
    const int brow = mtile * 16 + laneLo;
    const float* xrow = x + (size_t)brow * K_TOT;

    const int kbase = wave * K_PER_WAVE;

    v8f c = {0.f, 0.f, 0.f, 0.f, 0.f, 0.f, 0.f, 0.f};

    // prologue load (step 0)
    v2f af, bf;
    {
        const int ksel  = kbase + 2 * hiSel;
        const int wbase = ksel * OUT_DIM + laneLo;
        af.x = xrow[ksel];
        af.y = xrow[ksel + 1];
        bf.x = Wf[wbase];
        bf.y = Wf[wbase + OUT_DIM];
    }

#pragma unroll 4
    for (int s = 0; s < K_STEPS - 1; ++s) {
        // prefetch fragments for step s+1 before consuming step s
        const int ksel  = kbase + 4 * (s + 1) + 2 * hiSel;
        const int wbase = ksel * OUT_DIM + laneLo;
        v2f an, bn;
        an.x = xrow[ksel];
        an.y = xrow[ksel + 1];
        bn.x = Wf[wbase];
        bn.y = Wf[wbase + OUT_DIM];

        c = __builtin_amdgcn_wmma_f32_16x16x4_f32(
                false, af, false, bf, (short)0, c, false, false);
        af = an;
        bf = bn;
    }
    c = __builtin_amdgcn_wmma_f32_16x16x4_f32(
            false, af, false, bf, (short)0, c, false, false);

#pragma unroll
    for (int r = 0; r < 8; ++r) partials[wave][lane][r] = c[r];
    __syncthreads();

    // Fixed-order cross-wave reduction: thread t (<256) owns (lane l, vgpr r).
    if (tid < 256) {
        const int l = tid & 31;
        const int r = tid >> 5;
        float s = 0.f;
#pragma unroll
        for (int w = 0; w < WAVES; ++w) s += partials[w][l][r];
        // C/D layout: lanes 0-15 -> M = r, lanes 16-31 -> M = r + 8; N = l % 16
        const int row = r + 8 * (l >> 4);
        const int col = l & 15;
        smat[row][col] = s * (1.0f / (float)N_CAPS);   // mean (uniform c = 1/N)
    }
    __syncthreads();

    // squash per batch row, write v[64,16] to workspace
    if (tid < 256) {
        const int row = tid >> 4;
        const int col = tid & 15;
        float sq = 0.f;
#pragma unroll
        for (int d = 0; d < OUT_DIM; ++d) { float m = smat[row][d]; sq += m * m; }
        const float scale = sq / (1.0f + sq) / sqrtf(sq + 1e-8f);
        const int b = mtile * 16 + row;
        v_out[b * OUT_DIM + col] = scale * smat[row][col];
    }
}

// Kernel B: out[b, n, :] = v[b, :] broadcast over n, coalesced 128-bit stores.
// Grid covers exactly B*N*OUT_DIM/4 = 294912 float4 slots.
__global__ __launch_bounds__(256)
void caps_bcast_kernel(const float* __restrict__ v, float4* __restrict__ out) {
    const int q = blockIdx.x * blockDim.x + threadIdx.x;   // float4 slot
    const int per_b  = N_CAPS * (OUT_DIM / 4);             // 4608 slots per b
    const int b      = q / per_b;
    const int within = q - b * per_b;
    const int d4     = within & 3;                         // which quad of 16 dims
    const float4* vt = (const float4*)v;
    out[q] = vt[b * (OUT_DIM / 4) + d4];
}

extern "C" void kernel_launch(void* const* d_in, const int* in_sizes, int n_in,
                              void* d_out, int out_size, void* d_ws, size_t ws_size,
                              hipStream_t stream) {
    (void)in_sizes; (void)n_in; (void)out_size; (void)ws_size;
    const float* x  = (const float*)d_in[0];   // [64, 1152, 8]
    const float* Wf = (const float*)d_in[1];   // [1, 1152, 8, 16] -> flat [9216,16]
    float* v = (float*)d_ws;                   // [64,16] = 4 KB scratch

    caps_reduce_kernel<<<4, 1024, 0, stream>>>(x, Wf, v);

    const int total_f4 = B_DIM * N_CAPS * OUT_DIM / 4;     // 294912
    caps_bcast_kernel<<<total_f4 / 256, 256, 0, stream>>>(v, (float4*)d_out);
}